// NGCF_16527034155364
// MI455X (gfx1250) — compile-verified
//
#include <hip/hip_runtime.h>
#include <hip/hip_bf16.h>
#include <math.h>

#define U_CNT   8000
#define I_CNT   16000
#define N_NODES 24000
#define D_DIM   64
#define NNZ_CNT 768000
#define ALLD    192   // 3 * 64 concatenated embeddings
#define NTILES  1000  // I_CNT / 16

typedef float v2f __attribute__((ext_vector_type(2)));
typedef float v8f __attribute__((ext_vector_type(8)));

__device__ __forceinline__ float leaky(float x) { return x > 0.f ? x : 0.01f * x; }

// ---------------------------------------------------------------- gather ego0
__global__ void k_gather_ego(const int* __restrict__ uidx, const int* __restrict__ iidx,
                             const float* __restrict__ ue, const float* __restrict__ ie,
                             float* __restrict__ ego) {
  int t = blockIdx.x * blockDim.x + threadIdx.x;
  if (t >= N_NODES * D_DIM) return;
  int m = t >> 6, d = t & 63;
  ego[t] = (m < U_CNT) ? ue[uidx[m] * D_DIM + d]
                       : ie[iidx[m - U_CNT] * D_DIM + d];
}

__global__ void k_zero(float* __restrict__ p, int n) {
  int t = blockIdx.x * blockDim.x + threadIdx.x;
  if (t < n) p[t] = 0.f;
}

// ------------------------------------------------------- COO SpMM via atomics
__global__ void k_spmm(const int* __restrict__ rows, const int* __restrict__ cols,
                       const float* __restrict__ vals, const float* __restrict__ ego,
                       float* __restrict__ side) {
  long long t = (long long)blockIdx.x * blockDim.x + threadIdx.x;
  if (t >= (long long)NNZ_CNT * D_DIM) return;
  int e = (int)(t >> 6), d = (int)(t & 63);
  int r = rows[e], c = cols[e];
  atomicAdd(&side[r * D_DIM + d], vals[e] * ego[c * D_DIM + d]);
}

// ------------------------------------------------- fused propagation layer
// one wave per 16x16 tile; block of 4 waves covers the full N=64 width.
// sum = leaky(side @ gcW^T + gcb); bi = leaky((ego*side) @ biW^T + bib)
__global__ void __launch_bounds__(128)
k_layer(const float* __restrict__ side, const float* __restrict__ ego,
        const float* __restrict__ gcW, const float* __restrict__ gcb,
        const float* __restrict__ biW, const float* __restrict__ bib,
        float* __restrict__ ego_out) {
  int lane = threadIdx.x & 31;
  int wave = threadIdx.x >> 5;        // 0..3 -> N tile within D=64
  int mt   = blockIdx.x;              // 0..1499
  int mrow = mt * 16 + (lane & 15);   // A operand: lane holds row M
  int ncol = wave * 16 + (lane & 15); // B operand: lane holds col N
  int kgrp = (lane >> 4) * 2;         // K sub-offset per half-wave

  const float* aS = side + mrow * D_DIM;
  const float* aE = ego  + mrow * D_DIM;
  const float* bG = gcW  + ncol * D_DIM;   // gcW[n][k]  (GEMM vs W^T)
  const float* bB = biW  + ncol * D_DIM;

  v8f c_sum = {}; v8f c_bi = {};
#pragma unroll
  for (int k0 = 0; k0 < D_DIM; k0 += 4) {
    v2f as = *(const v2f*)(aS + k0 + kgrp);
    v2f ae = *(const v2f*)(aE + k0 + kgrp);
    v2f ab; ab.x = as.x * ae.x; ab.y = as.y * ae.y;
    v2f bg = *(const v2f*)(bG + k0 + kgrp);
    v2f bb = *(const v2f*)(bB + k0 + kgrp);
    c_sum = __builtin_amdgcn_wmma_f32_16x16x4_f32(false, as, false, bg,
                                                  (short)0, c_sum, false, false);
    c_bi  = __builtin_amdgcn_wmma_f32_16x16x4_f32(false, ab, false, bb,
                                                  (short)0, c_bi,  false, false);
  }
  float bgc = gcb[ncol], bbi = bib[ncol];
#pragma unroll
  for (int r = 0; r < 8; ++r) {
    int row = mt * 16 + r + 8 * (lane >> 4);  // C layout: M = vgpr + 8*(lane/16)
    ego_out[row * D_DIM + ncol] = leaky(c_sum[r] + bgc) + leaky(c_bi[r] + bbi);
  }
}

// ------------------------------------------------ pack [ego0|ego1|ego2] rows
__global__ void k_pack(const float* __restrict__ e0, const float* __restrict__ e1,
                       const float* __restrict__ e2, float* __restrict__ all) {
  int t = blockIdx.x * blockDim.x + threadIdx.x;
  if (t >= N_NODES * ALLD) return;
  int m = t / ALLD, k = t - m * ALLD;
  float v = (k < 64) ? e0[m * 64 + k]
          : (k < 128) ? e1[m * 64 + (k - 64)]
                      : e2[m * 64 + (k - 128)];
  all[t] = v;
}

// ----------------------------------------------------------------------------
// Pass A: per-row log-sum-exp of u_g @ i_g^T WITHOUT materializing scores.
// One block per 16-row stripe; 8 waves sweep all 1000 column tiles (125 each),
// keeping an online-softmax running (max, sumexp) per row, combined via LDS.
// ----------------------------------------------------------------------------
__global__ void __launch_bounds__(256)
k_row_lse(const float* __restrict__ all, float* __restrict__ lse) {
  __shared__ float sm[8][16][2];     // [wave][row-in-stripe][{max,sum}]
  int lane = threadIdx.x & 31;
  int wave = threadIdx.x >> 5;       // 0..7
  int mt   = blockIdx.x;             // 0..499
  int mrow = mt * 16 + (lane & 15);
  int kgrp = (lane >> 4) * 2;
  const float* A = all + mrow * ALLD;

  float runm[8], runs[8];
#pragma unroll
  for (int r = 0; r < 8; ++r) { runm[r] = -3.402823466e38f; runs[r] = 0.f; }

  for (int t = 0; t < NTILES / 8; ++t) {
    int nt   = wave + t * 8;                       // 0..999
    int nrow = nt * 16 + (lane & 15);
    const float* B = all + (U_CNT + nrow) * ALLD;
    __builtin_prefetch(B, 0, 0);                   // global_prefetch_b8
    __builtin_prefetch(B + 64, 0, 0);

    v8f c = {};
#pragma unroll
    for (int k0 = 0; k0 < ALLD; k0 += 4) {
      v2f a = *(const v2f*)(A + k0 + kgrp);
      v2f b = *(const v2f*)(B + k0 + kgrp);
      c = __builtin_amdgcn_wmma_f32_16x16x4_f32(false, a, false, b,
                                                (short)0, c, false, false);
    }
    // per-row reduction across the 16 lanes of each half-wave
    // (xor masks 1/2/4/8 never cross bit 4, so halves reduce independently)
#pragma unroll
    for (int r = 0; r < 8; ++r) {
      float v  = c[r];
      float mx = v;
      mx = fmaxf(mx, __shfl_xor(mx, 1));
      mx = fmaxf(mx, __shfl_xor(mx, 2));
      mx = fmaxf(mx, __shfl_xor(mx, 4));
      mx = fmaxf(mx, __shfl_xor(mx, 8));
      float s = expf(v - mx);
      s += __shfl_xor(s, 1);
      s += __shfl_xor(s, 2);
      s += __shfl_xor(s, 4);
      s += __shfl_xor(s, 8);
      float nm = fmaxf(runm[r], mx);
      runs[r]  = runs[r] * expf(runm[r] - nm) + s * expf(mx - nm);
      runm[r]  = nm;
    }
  }

  if ((lane & 15) == 0) {
    int h = lane >> 4;
#pragma unroll
    for (int r = 0; r < 8; ++r) {
      sm[wave][r + 8 * h][0] = runm[r];
      sm[wave][r + 8 * h][1] = runs[r];
    }
  }
  __syncthreads();
  if (threadIdx.x < 16) {
    int row = threadIdx.x;
    float M = -3.402823466e38f;
    for (int w = 0; w < 8; ++w) M = fmaxf(M, sm[w][row][0]);
    float S = 0.f;
    for (int w = 0; w < 8; ++w) S += sm[w][row][1] * expf(sm[w][row][0] - M);
    lse[mt * 16 + row] = M + logf(S);
  }
}

// ----------------------------------------------------------------------------
// Pass B: recompute scores (operands are L2-resident) and stream out
// log_softmax = score - lse[row] in a single 512 MB write.
// ----------------------------------------------------------------------------
__global__ void __launch_bounds__(256)
k_scores_out(const float* __restrict__ all, const float* __restrict__ lse,
             float* __restrict__ out) {
  int lane = threadIdx.x & 31;
  int wave = threadIdx.x >> 5;            // 0..7
  int nt   = blockIdx.x * 8 + wave;       // 0..999 item tiles
  int mt   = blockIdx.y;                  // 0..499  user tiles
  int mrow = mt * 16 + (lane & 15);
  int nrow = nt * 16 + (lane & 15);
  int kgrp = (lane >> 4) * 2;

  const float* A = all + mrow * ALLD;
  const float* B = all + (U_CNT + nrow) * ALLD;
  __builtin_prefetch(B, 0, 0);
  __builtin_prefetch(B + 64, 0, 0);

  v8f c = {};
#pragma unroll
  for (int k0 = 0; k0 < ALLD; k0 += 4) {
    v2f a = *(const v2f*)(A + k0 + kgrp);
    v2f b = *(const v2f*)(B + k0 + kgrp);
    c = __builtin_amdgcn_wmma_f32_16x16x4_f32(false, a, false, b,
                                              (short)0, c, false, false);
  }
#pragma unroll
  for (int r = 0; r < 8; ++r) {
    int row = mt * 16 + r + 8 * (lane >> 4);
    out[(size_t)row * I_CNT + nt * 16 + (lane & 15)] = c[r] - lse[row];
  }
}

// ---------------------------------------------------------------- launcher
extern "C" void kernel_launch(void* const* d_in, const int* in_sizes, int n_in,
                              void* d_out, int out_size, void* d_ws, size_t ws_size,
                              hipStream_t stream) {
  const int*   uidx  = (const int*)d_in[0];
  const int*   iidx  = (const int*)d_in[1];
  const int*   arows = (const int*)d_in[2];
  const int*   acols = (const int*)d_in[3];
  const float* avals = (const float*)d_in[4];
  const float* ue    = (const float*)d_in[5];
  const float* ie    = (const float*)d_in[6];
  const float* gcW   = (const float*)d_in[7];
  const float* gcb   = (const float*)d_in[8];
  const float* biW   = (const float*)d_in[9];
  const float* bib   = (const float*)d_in[10];

  float* ws = (float*)d_ws;
  const size_t EG = (size_t)N_NODES * D_DIM;
  float* ego0 = ws;
  float* ego1 = ego0 + EG;
  float* ego2 = ego1 + EG;
  float* side = ego2 + EG;
  float* all  = side + EG;
  float* lse  = all + (size_t)N_NODES * ALLD;   // 8000 floats
  float* out  = (float*)d_out;

  {
    int n = N_NODES * D_DIM;
    k_gather_ego<<<(n + 255) / 256, 256, 0, stream>>>(uidx, iidx, ue, ie, ego0);
  }

  const float* ein[2]  = {ego0, ego1};
  float*       eout[2] = {ego1, ego2};
  for (int l = 0; l < 2; ++l) {
    k_zero<<<(N_NODES * D_DIM + 255) / 256, 256, 0, stream>>>(side, N_NODES * D_DIM);
    long long tot = (long long)NNZ_CNT * D_DIM;
    k_spmm<<<(int)((tot + 255) / 256), 256, 0, stream>>>(arows, acols, avals, ein[l], side);
    k_layer<<<N_NODES / 16, 128, 0, stream>>>(
        side, ein[l],
        gcW + (size_t)l * D_DIM * D_DIM, gcb + (size_t)l * D_DIM,
        biW + (size_t)l * D_DIM * D_DIM, bib + (size_t)l * D_DIM,
        eout[l]);
  }

  k_pack<<<(N_NODES * ALLD + 255) / 256, 256, 0, stream>>>(ego0, ego1, ego2, all);

  // Pass A: per-row logsumexp, no score materialization
  k_row_lse<<<U_CNT / 16, 256, 0, stream>>>(all, lse);

  // Pass B: recompute + single streamed write of log-softmax
  dim3 g(NTILES / 8, U_CNT / 16);   // 125 x 500
  k_scores_out<<<g, 256, 0, stream>>>(all, lse, out);
}